// Kron_MLP_77223511982250
// MI455X (gfx1250) — compile-verified
//
#include <hip/hip_runtime.h>
#include <math.h>

// Problem constants (from reference)
#define NPTS 512
#define MPTS 512
#define W0   256
#define WID  256
#define ODIM 3

typedef __attribute__((ext_vector_type(2))) float v2f;
typedef __attribute__((ext_vector_type(4))) float v4f;
typedef __attribute__((ext_vector_type(8))) float v8f;

// LDS: W1 repacked into WMMA-B fragment order (256 KB) + u[w][2] (2 KB)
#define LDS_B_FLOATS (W0 * WID)                        // 65536 floats
#define LDS_TOTAL_BYTES ((LDS_B_FLOATS + 2 * W0) * 4)  // 264192 B <= 320 KB/WGP

// One workgroup per n. 256 threads = 8 wave32.
// Wave handles m-tiles {wave, wave+8, wave+16, wave+24} (16 rows each).
// Inner GEMM: [16 m x 256 w] @ [256 w x 256 k] via v_wmma_f32_16x16x4_f32.
__global__ __launch_bounds__(256) void kron_mlp_fused(
    const float* __restrict__ x, const float* __restrict__ y,
    const float* __restrict__ A, const float* __restrict__ W1,
    const float* __restrict__ b1, const float* __restrict__ W2,
    const float* __restrict__ b2, float* __restrict__ out)
{
    extern __shared__ float smem[];
    float* Blds = smem;                 // fragment-packed W1
    float* Ulds = smem + LDS_B_FLOATS;  // u[w] as interleaved float2

    const int tid  = threadIdx.x;
    const int n    = blockIdx.x;
    const int wave = tid >> 5;
    const int lane = tid & 31;
    const int lh   = lane >> 4;  // which 16-lane half
    const int lr   = lane & 15;

    // ---- Phase 0a: u[w][j] = sum_i x[n,i] * A[w,i,j]  (tid == w) ----
    {
        const v4f a4 = ((const v4f*)A)[tid];  // A[w,0,0],A[w,0,1],A[w,1,0],A[w,1,1]
        const float x0 = x[2 * n + 0], x1 = x[2 * n + 1];
        Ulds[2 * tid + 0] = x0 * a4[0] + x1 * a4[2];
        Ulds[2 * tid + 1] = x0 * a4[1] + x1 * a4[3];
    }

    // ---- Phase 0b: repack W1[k,w] into per-lane WMMA-B fragment order ----
    // Done with CDNA5 async global->LDS scatter: the DMA writes each element
    // straight to its permuted LDS slot (no VGPR data, no ds_store). Global
    // side is fully coalesced (lanes read consecutive w of one W1 row).
    // B element for (kt, s): lane = (q>>1)*16 + r, comp c = q&1
    //   holds W1[kt*16 + r, s*4 + q]   (B[Kw][Nk] = W1^T fragment)
    {
        const uint32_t blds_base = (uint32_t)(uintptr_t)Blds;  // LDS byte addr = low 32 bits
        for (int idx = tid; idx < W0 * WID; idx += 256) {
            const int k = idx >> 8, w = idx & 255;
            const int kt = k >> 4, r = k & 15, s = w >> 2, q = w & 3;
            const int off = (((kt * 64 + s) * 32) + ((q >> 1) * 16 + r)) * 2 + (q & 1);
            const uint32_t lds_byte = blds_base + (uint32_t)off * 4u;
            const uint32_t g_byte   = (uint32_t)idx * 4u;
            asm volatile("global_load_async_to_lds_b32 %0, %1, %2"
                         :
                         : "v"(lds_byte), "v"(g_byte), "s"(W1)
                         : "memory");
        }
        asm volatile("s_wait_asynccnt 0" ::: "memory");
    }
    __syncthreads();

    for (int mi = 0; mi < 4; ++mi) {
        const int mt = wave + 8 * mi;

        // ---- A fragments: H[m, w] = relu(y[m,0]*u[w,0] + y[m,1]*u[w,1]) ----
        // A layout (16x4 f32): lanes 0-15 -> M=lr, K=0,1 ; lanes 16-31 -> M=lr, K=2,3
        const int mrow = mt * 16 + lr;
        const float y0 = y[2 * mrow + 0], y1 = y[2 * mrow + 1];
        v2f Af[64];  // 128 VGPRs, reused across all 16 k-tiles
        #pragma unroll
        for (int s = 0; s < 64; ++s) {
            // w = s*4 + 2*lh + {0,1}; u pairs are contiguous float2s
            const v4f uq = *(const v4f*)(Ulds + s * 8 + lh * 4);
            Af[s][0] = fmaxf(y0 * uq[0] + y1 * uq[1], 0.0f);
            Af[s][1] = fmaxf(y0 * uq[2] + y1 * uq[3], 0.0f);
        }

        // Per-lane partials for the 3-wide head (rows j + 8*lh)
        float op[8][3];
        #pragma unroll
        for (int j = 0; j < 8; ++j) { op[j][0] = 0.f; op[j][1] = 0.f; op[j][2] = 0.f; }

        for (int kt = 0; kt < 16; ++kt) {
            // Two independent accumulator chains (even/odd step) for XDL pipelining
            v8f acc0 = {0.f, 0.f, 0.f, 0.f, 0.f, 0.f, 0.f, 0.f};
            v8f acc1 = {0.f, 0.f, 0.f, 0.f, 0.f, 0.f, 0.f, 0.f};
            const float* bbase = Blds + kt * 4096 + lane * 2;

            // Software-pipelined B fragments: groups of 8, ping-pong buffers.
            v2f buf[2][8];
            #pragma unroll
            for (int t = 0; t < 8; ++t)
                buf[0][t] = *(const v2f*)(bbase + t * 64);

            #pragma unroll
            for (int g = 0; g < 8; ++g) {
                if (g < 7) {  // prefetch next group while this one computes
                    #pragma unroll
                    for (int t = 0; t < 8; ++t)
                        buf[(g + 1) & 1][t] = *(const v2f*)(bbase + ((g + 1) * 8 + t) * 64);
                }
                #pragma unroll
                for (int t = 0; t < 8; ++t) {
                    if ((t & 1) == 0)
                        acc0 = __builtin_amdgcn_wmma_f32_16x16x4_f32(
                            false, Af[g * 8 + t], false, buf[g & 1][t], (short)0,
                            acc0, false, false);
                    else
                        acc1 = __builtin_amdgcn_wmma_f32_16x16x4_f32(
                            false, Af[g * 8 + t], false, buf[g & 1][t], (short)0,
                            acc1, false, false);
                }
            }

            // C/D layout: VGPR j, lanes 0-15 -> (M=j, N=lr); lanes 16-31 -> (M=8+j, N=lr)
            const int kcol = kt * 16 + lr;
            const float b1v = b1[kcol];
            const float w2a = W2[0 * WID + kcol];
            const float w2b = W2[1 * WID + kcol];
            const float w2c = W2[2 * WID + kcol];
            #pragma unroll
            for (int j = 0; j < 8; ++j) {
                const float g = fmaxf((acc0[j] + acc1[j]) + b1v, 0.0f);  // ReLU(H@W1^T + b1)
                op[j][0] += g * w2a;
                op[j][1] += g * w2b;
                op[j][2] += g * w2c;
            }
        }

        // Reduce the k-dimension across the 16 lanes of each half
        #pragma unroll
        for (int t = 1; t < 16; t <<= 1) {
            #pragma unroll
            for (int j = 0; j < 8; ++j) {
                op[j][0] += __shfl_xor(op[j][0], t, 32);
                op[j][1] += __shfl_xor(op[j][1], t, 32);
                op[j][2] += __shfl_xor(op[j][2], t, 32);
            }
        }
        if (lr == 0) {  // lanes 0 (rows 0..7) and 16 (rows 8..15) write
            #pragma unroll
            for (int j = 0; j < 8; ++j) {
                const int m = mt * 16 + 8 * lh + j;
                float* o = out + (size_t)(n * MPTS + m) * ODIM;
                #pragma unroll
                for (int c = 0; c < ODIM; ++c) {
                    const float z = op[j][c] + b2[c];
                    o[c] = 1.0f / (1.0f + __expf(-z));  // sigmoid
                }
            }
        }
    }
}

extern "C" void kernel_launch(void* const* d_in, const int* in_sizes, int n_in,
                              void* d_out, int out_size, void* d_ws, size_t ws_size,
                              hipStream_t stream) {
    const float* x  = (const float*)d_in[0];
    const float* y  = (const float*)d_in[1];
    const float* A  = (const float*)d_in[2];
    const float* W1 = (const float*)d_in[3];
    const float* b1 = (const float*)d_in[4];
    const float* W2 = (const float*)d_in[5];
    const float* b2 = (const float*)d_in[6];
    float* out = (float*)d_out;

    // Allow 258 KB of dynamic LDS (CDNA5 WGP has 320 KB). Idempotent; not a
    // stream op, safe under graph capture.
    (void)hipFuncSetAttribute((const void*)kron_mlp_fused,
                              hipFuncAttributeMaxDynamicSharedMemorySize,
                              LDS_TOTAL_BYTES);

    kron_mlp_fused<<<NPTS, 256, LDS_TOTAL_BYTES, stream>>>(x, y, A, W1, b1, W2, b2, out);
}